// SAFLCrowdLoss_39702677684512
// MI455X (gfx1250) — compile-verified
//
#include <hip/hip_runtime.h>
#include <math.h>

// ---------------------------------------------------------------------------
// Problem constants (reference: B=32, H=W=768, LEVELS=(1,2,3) -> finest 8x8)
// ---------------------------------------------------------------------------
#define BATCH     32
#define HW        768
#define GRID8     8            // 8x8 cells at finest level
#define CELLPX    96           // 768 / 8
#define NCELL     64           // cells per image
#define NBLK      (BATCH * NCELL)      // 2048 blocks in pass 1
#define VEC_PER_CELL 2304      // 96*96/4 float4 per cell per buffer
#define ROWV      192          // 768/4 float4 per image row
#define CELLROWV  24           // 96/4 float4 per cell row

typedef __attribute__((ext_vector_type(2))) float v2f;
typedef __attribute__((ext_vector_type(8))) float v8f;

// ---------------------------------------------------------------------------
// Exact f32 wave32 sum via V_WMMA_F32_16X16X4_F32.
// A = {x, 0} per lane, B = all-ones. D[m][n] = rowsum_m(A) independent of the
// A/B lane layouts (every lane's x occupies exactly one A slot, B is 1 in all
// 64 slots). C/D layout: lanes 0-15 hold D rows 0-7,
// lanes 16-31 hold rows 8-15, one row per dest VGPR. Sum of the 8 components
// gives half the row sums; shfl_xor(16) adds the other half -> full Σx.
// EXEC must be all ones at the call site (callers are fully convergent).
// ---------------------------------------------------------------------------
__device__ __forceinline__ float wave_sum_wmma(float x) {
    v2f a; a[0] = x;    a[1] = 0.0f;
    v2f b; b[0] = 1.0f; b[1] = 1.0f;
    v8f c = {};
    c = __builtin_amdgcn_wmma_f32_16x16x4_f32(
            /*neg_a=*/false, a, /*neg_b=*/false, b,
            /*c_mod=*/(short)0, c, /*reuse_a=*/false, /*reuse_b=*/false);
    float s = ((c[0] + c[1]) + (c[2] + c[3])) + ((c[4] + c[5]) + (c[6] + c[7]));
    s += __shfl_xor(s, 16, 32);
    return s;  // broadcast to all 32 lanes
}

// ---------------------------------------------------------------------------
// Pass 1: one block per (batch, 8x8 cell). Streams the 96x96 tile of pred and
// gt once with b128 loads, producing: cell sum of pred, cell sum of gt, and
// the block-local sum of squared differences.
// ---------------------------------------------------------------------------
__global__ __launch_bounds__(256)
void cell_reduce_kernel(const float* __restrict__ pred,
                        const float* __restrict__ gt,
                        float* __restrict__ cellP,   // [NCELL][BATCH] transposed
                        float* __restrict__ cellG,   // [NCELL][BATCH] transposed
                        float* __restrict__ ssdPart) // [NBLK]
{
    const int blk  = blockIdx.x;          // 0..2047
    const int b    = blk >> 6;            // batch
    const int cell = blk & 63;            // 0..63
    const int cy   = cell >> 3;
    const int cx   = cell & 7;
    const int t    = threadIdx.x;

    const size_t base = ((size_t)b * HW + (size_t)cy * CELLPX) * HW + (size_t)cx * CELLPX;
    const float4* __restrict__ p4 = (const float4*)(pred + base);
    const float4* __restrict__ g4 = (const float4*)(gt   + base);

    float sp = 0.0f, sg = 0.0f, ss = 0.0f;
#pragma unroll
    for (int i = 0; i < VEC_PER_CELL / 256; ++i) {       // 9 iterations
        const int idx = t + i * 256;                     // 0..2303
        const int row = idx / CELLROWV;                  // 0..95
        const int c4  = idx - row * CELLROWV;            // 0..23
        const size_t off = (size_t)row * ROWV + (size_t)c4;
        const float4 p = p4[off];
        const float4 g = g4[off];
        sp += (p.x + p.y) + (p.z + p.w);
        sg += (g.x + g.y) + (g.z + g.w);
        const float dx = p.x - g.x, dy = p.y - g.y;
        const float dz = p.z - g.z, dw = p.w - g.w;
        ss = fmaf(dx, dx, ss); ss = fmaf(dy, dy, ss);
        ss = fmaf(dz, dz, ss); ss = fmaf(dw, dw, ss);
    }

    // Intra-wave reduction on the WMMA path (exact f32).
    const float P = wave_sum_wmma(sp);
    const float G = wave_sum_wmma(sg);
    const float S = wave_sum_wmma(ss);

    __shared__ float redP[8], redG[8], redS[8];
    const int wave = t >> 5;
    const int lane = t & 31;
    if (lane == 0) { redP[wave] = P; redG[wave] = G; redS[wave] = S; }
    __syncthreads();
    if (t == 0) {
        float tp = 0.0f, tg = 0.0f, ts = 0.0f;
#pragma unroll
        for (int w = 0; w < 8; ++w) { tp += redP[w]; tg += redG[w]; ts += redS[w]; }
        cellP[cell * BATCH + b] = tp;   // transposed: lane-coalesced in pass 2
        cellG[cell * BATCH + b] = tg;
        ssdPart[blk]            = ts;
    }
}

// ---------------------------------------------------------------------------
// Pass 2: single block. Reduces SSD partials; lane b (wave 0) derives the
// count / regional(L1,L2,L3) / domain-CE terms for batch b from the 64 cell
// sums; shuffle-reduce across 32 lanes; thread 0 writes the weighted total.
// ---------------------------------------------------------------------------
__global__ __launch_bounds__(256)
void finalize_kernel(const float* __restrict__ cellP,
                     const float* __restrict__ cellG,
                     const float* __restrict__ ssdPart,
                     const float* __restrict__ dom_rgb,   // [BATCH][2]
                     const float* __restrict__ dom_th,    // [BATCH][2]
                     float* __restrict__ out)
{
    const int t    = threadIdx.x;
    const int wave = t >> 5;
    const int lane = t & 31;
    __shared__ float lds[8];

    // ---- Σ(p-g)^2 over all elements: reduce 2048 partials ------------------
    float s = 0.0f;
    for (int i = t; i < NBLK; i += 256) s += ssdPart[i];
#pragma unroll
    for (int o = 16; o > 0; o >>= 1) s += __shfl_xor(s, o, 32);
    if (lane == 0) lds[wave] = s;
    __syncthreads();
    float ssd_total = 0.0f;
    if (t == 0) {
#pragma unroll
        for (int w = 0; w < 8; ++w) ssd_total += lds[w];
    }

    // ---- per-batch terms on wave 0 ----------------------------------------
    if (t < 32) {
        const int b = t;
        float d[NCELL];
        float sumP = 0.0f, sumG = 0.0f;
#pragma unroll
        for (int i = 0; i < NCELL; ++i) {
            const float P = cellP[i * BATCH + b];   // coalesced across lanes
            const float G = cellG[i * BATCH + b];
            d[i] = P - G;
            sumP += P; sumG += G;
        }
        const float cl = fabsf(sumP - sumG);        // per-batch |count diff|

        float reg = 0.0f;
        // Level 3: 8x8 cells
#pragma unroll
        for (int i = 0; i < NCELL; ++i) reg += fabsf(d[i]);
        // Level 2: 4x4 cells = 2x2 blocks of level-3 cells
#pragma unroll
        for (int Y = 0; Y < 4; ++Y)
#pragma unroll
            for (int X = 0; X < 4; ++X) {
                const float a = d[(2 * Y) * 8 + 2 * X]     + d[(2 * Y) * 8 + 2 * X + 1]
                              + d[(2 * Y + 1) * 8 + 2 * X] + d[(2 * Y + 1) * 8 + 2 * X + 1];
                reg += fabsf(a);
            }
        // Level 1: 2x2 cells = 4x4 blocks of level-3 cells
#pragma unroll
        for (int Y = 0; Y < 2; ++Y)
#pragma unroll
            for (int X = 0; X < 2; ++X) {
                float a = 0.0f;
#pragma unroll
                for (int yy = 0; yy < 4; ++yy)
#pragma unroll
                    for (int xx = 0; xx < 4; ++xx)
                        a += d[(4 * Y + yy) * 8 + 4 * X + xx];
                reg += fabsf(a);
            }

        // Domain cross-entropy (labels: rgb=0, thermal=1)
        const float r0 = dom_rgb[2 * b], r1 = dom_rgb[2 * b + 1];
        const float mr = fmaxf(r0, r1);
        const float ce_rgb = (mr + logf(expf(r0 - mr) + expf(r1 - mr))) - r0;
        const float t0 = dom_th[2 * b], t1 = dom_th[2 * b + 1];
        const float mt = fmaxf(t0, t1);
        const float ce_th = (mt + logf(expf(t0 - mt) + expf(t1 - mt))) - t1;

        // Reduce the four per-batch quantities across the 32 lanes.
        float cls = cl, regs = reg, cr = ce_rgb, ct = ce_th;
#pragma unroll
        for (int o = 16; o > 0; o >>= 1) {
            cls  += __shfl_xor(cls,  o, 32);
            regs += __shfl_xor(regs, o, 32);
            cr   += __shfl_xor(cr,   o, 32);
            ct   += __shfl_xor(ct,   o, 32);
        }

        if (t == 0) {
            const float inv_b    = 1.0f / (float)BATCH;
            const float density  = ssd_total / ((float)BATCH * (float)HW * (float)HW);
            const float count_l  = cls * inv_b;
            const float regional = (regs * inv_b) / (3.0f * 64.0f);  // LEVELS=3, last 8x8
            const float domain   = 0.5f * (cr * inv_b + ct * inv_b);
            out[0] = 100.0f * density + 0.001f * count_l
                   + 1.0f * regional + 0.5f * domain;
        }
    }
}

// ---------------------------------------------------------------------------
// Launcher
// ---------------------------------------------------------------------------
extern "C" void kernel_launch(void* const* d_in, const int* in_sizes, int n_in,
                              void* d_out, int out_size, void* d_ws, size_t ws_size,
                              hipStream_t stream) {
    const float* pred    = (const float*)d_in[0];  // [32,1,768,768]
    const float* gt      = (const float*)d_in[1];  // [32,1,768,768]
    const float* dom_rgb = (const float*)d_in[2];  // [32,2]
    const float* dom_th  = (const float*)d_in[3];  // [32,2]
    float*       out     = (float*)d_out;

    float* ws      = (float*)d_ws;
    float* cellP   = ws;                 // NBLK floats
    float* cellG   = ws + NBLK;          // NBLK floats
    float* ssdPart = ws + 2 * NBLK;      // NBLK floats   (24 KB total)

    cell_reduce_kernel<<<NBLK, 256, 0, stream>>>(pred, gt, cellP, cellG, ssdPart);
    finalize_kernel<<<1, 256, 0, stream>>>(cellP, cellG, ssdPart, dom_rgb, dom_th, out);
}